// Memory_44461501448361
// MI455X (gfx1250) — compile-verified
//
#include <hip/hip_runtime.h>
#include <hip/hip_bf16.h>

typedef __attribute__((ext_vector_type(2))) float v2f;
typedef __attribute__((ext_vector_type(8))) float v8f;

#define NPTS   500000
#define FDIM   256
#define MQ     1024
#define QBLK   (MQ / 16)          // 64 query blocks of 16 rows
#define NSLICE 8
#define NSL    ((NPTS + NSLICE - 1) / NSLICE)   // 62500 per slice

// ---- workspace layout (bytes) ----
static constexpr size_t OFF_PVAL  = 0;              // 64*8*16 floats
static constexpr size_t OFF_PIDX  = 32768;          // 64*8*16 ints
static constexpr size_t OFF_H1    = 65536;          // 1024 ints
static constexpr size_t OFF_H2    = 69632;          // 512 ints
static constexpr size_t OFF_META  = 71680;          // 8 ints: c, base, needed, subH, counter
static constexpr size_t OFF_BUF   = 71712;          // 4096 ints (candidate keys)
static constexpr size_t OFF_EVICT = 88096;          // 1024 ints
static constexpr size_t OFF_DELTA = 92192;          // NPTS ints (usage scatter deltas)

// ============================================================================
// Kernel 1: nearest-neighbor via V_WMMA_F32_16X16X4_F32.
// A row m = (-2qx, -2qy, -2qz, 1); B col n = (px, py, pz, |p|^2)
//   => D[m][n] = |p_n|^2 - 2 q_m . p_n   (argmin-equivalent to distance)
// ============================================================================
__global__ void nn_wmma_kernel(const float* __restrict__ points,
                               const float* __restrict__ mem_points,
                               float* __restrict__ pval, int* __restrict__ pidx) {
  const int lane  = threadIdx.x & 31;
  const int wave  = threadIdx.x >> 5;
  const int qb    = blockIdx.x;
  const int slice = blockIdx.y;
  const int col16 = lane & 15;

  // ---- A operand: 16x4 f32, lanes 0-15 hold K=0,1; lanes 16-31 hold K=2,3
  const int mrow = qb * 16 + col16;
  const float qx = points[mrow * 3 + 0];
  const float qy = points[mrow * 3 + 1];
  const float qz = points[mrow * 3 + 2];
  v2f a;
  if (lane < 16) { a[0] = -2.0f * qx; a[1] = -2.0f * qy; }
  else           { a[0] = -2.0f * qz; a[1] = 1.0f;       }

  const int n0base  = slice * NSL;
  const int nEnd    = (n0base + NSL < NPTS) ? (n0base + NSL) : NPTS;
  const int nchunks = (nEnd - n0base + 15) >> 4;

  float best[8];
  int   bidx[8];
#pragma unroll
  for (int v = 0; v < 8; ++v) { best[v] = 3.4e38f; bidx[v] = 0; }

  for (int t = wave; t < nchunks; t += 8) {
    const int col = n0base + t * 16 + col16;
    const int cc  = (col < NPTS) ? col : (NPTS - 1);
    const float px = mem_points[cc * 3 + 0];
    const float py = mem_points[cc * 3 + 1];
    const float pz = mem_points[cc * 3 + 2];
    const float pn2 = px * px + py * py + pz * pz;
    v2f b;
    if (lane < 16) { b[0] = px; b[1] = py;  }
    else           { b[0] = pz; b[1] = pn2; }
    v8f c = {0.f, 0.f, 0.f, 0.f, 0.f, 0.f, 0.f, 0.f};
    c = __builtin_amdgcn_wmma_f32_16x16x4_f32(false, a, false, b, (short)0, c,
                                              false, false);
    const bool ok = col < nEnd;
#pragma unroll
    for (int v = 0; v < 8; ++v) {
      const float s = c[v];
      const bool take = ok && (s < best[v]);
      best[v] = take ? s   : best[v];
      bidx[v] = take ? col : bidx[v];
    }
  }

  // intra-half (16-lane) reduction; xor masks 1..8 stay within each half
#pragma unroll
  for (int off = 1; off <= 8; off <<= 1) {
#pragma unroll
    for (int v = 0; v < 8; ++v) {
      const float ov = __shfl_xor(best[v], off, 32);
      const int   oi = __shfl_xor(bidx[v], off, 32);
      const bool take = (ov < best[v]) || (ov == best[v] && oi < bidx[v]);
      best[v] = take ? ov : best[v];
      bidx[v] = take ? oi : bidx[v];
    }
  }

  __shared__ float sval[8][16];
  __shared__ int   sidx[8][16];
  if (lane == 0) {
#pragma unroll
    for (int v = 0; v < 8; ++v) { sval[wave][v] = best[v]; sidx[wave][v] = bidx[v]; }
  } else if (lane == 16) {
#pragma unroll
    for (int v = 0; v < 8; ++v) { sval[wave][8 + v] = best[v]; sidx[wave][8 + v] = bidx[v]; }
  }
  __syncthreads();
  if (threadIdx.x < 16) {
    const int r = threadIdx.x;
    float bv = sval[0][r]; int bi = sidx[0][r];
    for (int w = 1; w < 8; ++w) {
      const float ov = sval[w][r]; const int oi = sidx[w][r];
      const bool take = (ov < bv) || (ov == bv && oi < bi);
      bv = take ? ov : bv; bi = take ? oi : bi;
    }
    const int o = (qb * NSLICE + slice) * 16 + r;
    pval[o] = bv; pidx[o] = bi;
  }
}

// ============================================================================
// Eviction-list construction (stable top-M least-used)
// ============================================================================
__global__ void hist1_kernel(const int* __restrict__ usage, int* __restrict__ h1) {
  const int i = blockIdx.x * blockDim.x + threadIdx.x;
  if (i < NPTS) {
    int u = usage[i]; u = u < 0 ? 0 : (u > 1023 ? 1023 : u);
    atomicAdd(&h1[u], 1);
  }
}

__global__ void scan1_kernel(const int* __restrict__ h1, int* __restrict__ meta) {
  __shared__ int h[1024];
  const int t = threadIdx.x;
  h[t] = h1[t];
  __syncthreads();
  for (int off = 1; off < 1024; off <<= 1) {
    const int add = (t >= off) ? h[t - off] : 0;
    __syncthreads();
    h[t] += add;
    __syncthreads();
  }
  const int cum  = h[t];
  const int prev = t ? h[t - 1] : 0;
  if (cum >= MQ && prev < MQ) {
    meta[0] = t;         // cutoff usage value c
    meta[1] = prev;      // count of usage < c
    meta[2] = MQ - prev; // needed from bin c
  }
}

__global__ void hist2_kernel(const int* __restrict__ usage,
                             const int* __restrict__ meta, int* __restrict__ h2) {
  const int c = meta[0];
  const int i = blockIdx.x * blockDim.x + threadIdx.x;
  if (i < NPTS) {
    int u = usage[i]; u = u < 0 ? 0 : (u > 1023 ? 1023 : u);
    if (u == c) atomicAdd(&h2[i >> 10], 1);
  }
}

__global__ void scan2_kernel(const int* __restrict__ h2, int* __restrict__ meta) {
  __shared__ int h[512];
  const int t = threadIdx.x;
  h[t] = h2[t];
  __syncthreads();
  for (int off = 1; off < 512; off <<= 1) {
    const int add = (t >= off) ? h[t - off] : 0;
    __syncthreads();
    h[t] += add;
    __syncthreads();
  }
  const int needed = meta[2];
  const int cum  = h[t];
  const int prev = t ? h[t - 1] : 0;
  if (cum >= needed && prev < needed) meta[3] = t;  // cutoff sub-bin
}

__global__ void collect_kernel(const int* __restrict__ usage, int* __restrict__ meta,
                               int* __restrict__ buf) {
  const int c = meta[0], subH = meta[3];
  const int i = blockIdx.x * blockDim.x + threadIdx.x;
  if (i >= NPTS) return;
  int u = usage[i]; u = u < 0 ? 0 : (u > 1023 ? 1023 : u);
  const bool sel = (u < c) || (u == c && (i >> 10) <= subH);
  if (sel) {
    const int pos = atomicAdd(&meta[4], 1);
    if (pos < 4096) buf[pos] = (u << 19) | i;   // distinct composite keys
  }
}

__global__ void select_kernel(const int* __restrict__ meta, const int* __restrict__ buf,
                              int* __restrict__ evict) {
  __shared__ int k[4096];
  int n = meta[4]; if (n > 4096) n = 4096;
  for (int j = threadIdx.x; j < 4096; j += 1024) k[j] = (j < n) ? buf[j] : 0x7FFFFFFF;
  __syncthreads();
  for (int j = threadIdx.x; j < n; j += 1024) {
    const int key = k[j];
    int r = 0;
    for (int t = 0; t < n; ++t) r += (k[t] < key) ? 1 : 0;   // keys distinct -> exact rank
    if (r < MQ) evict[r] = key & 0x7FFFF;
  }
}

// ============================================================================
// Finalize queries: reduce slice partials, mask, block-scan for ranks,
// emit idx output (as float) and usage scatter deltas.
// ============================================================================
__global__ void finalize_kernel(const float* __restrict__ points,
                                const float* __restrict__ pval,
                                const int* __restrict__ pidx,
                                const int* __restrict__ evict,
                                float* __restrict__ out_idx, int* __restrict__ delta) {
  __shared__ int scan[MQ];
  const int m = threadIdx.x;
  const int qb = m >> 4, r = m & 15;
  float bv = 3.4e38f; int bi = 0;
  for (int s = 0; s < NSLICE; ++s) {
    const int o = (qb * NSLICE + s) * 16 + r;
    const float ov = pval[o]; const int oi = pidx[o];
    const bool take = (ov < bv) || (ov == bv && oi < bi);
    bv = take ? ov : bv; bi = take ? oi : bi;
  }
  const float qx = points[m * 3 + 0], qy = points[m * 3 + 1], qz = points[m * 3 + 2];
  const float d2 = qx * qx + qy * qy + qz * qz + bv;  // |q|^2 + (|p|^2 - 2qp)
  const int mask = (d2 > 1e-6f) ? 1 : 0;              // dmin > 1e-3
  scan[m] = mask;
  __syncthreads();
  for (int off = 1; off < MQ; off <<= 1) {
    const int add = (m >= off) ? scan[m - off] : 0;
    __syncthreads();
    scan[m] += add;
    __syncthreads();
  }
  const int rank = scan[m] - 1;
  const int fin = mask ? evict[rank] : bi;
  out_idx[m] = (float)fin;
  atomicAdd(&delta[fin], 1);
}

// ============================================================================
// Output writers
// ============================================================================
__global__ void usage_out_kernel(const int* __restrict__ usage,
                                 const int* __restrict__ delta,
                                 float* __restrict__ out_usage) {
  const int i = blockIdx.x * blockDim.x + threadIdx.x;
  if (i < NPTS) out_usage[i] = (float)(usage[i] + delta[i]);
}

// out[N, 259] = concat(mem_points, mem_descriptors); one wave per row
__global__ void copy_out_kernel(const float* __restrict__ mem_points,
                                const float* __restrict__ mem_desc,
                                float* __restrict__ out) {
  const int wave = threadIdx.x >> 5, lane = threadIdx.x & 31;
  const long long row = (long long)blockIdx.x * 8 + wave;
  if (row >= NPTS) return;
  const float4* drow = (const float4*)(mem_desc + row * FDIM);  // 16B-aligned
  float* orow = out + row * (FDIM + 3);
  const float4 A = drow[lane * 2];
  const float4 B = drow[lane * 2 + 1];
  float* o = orow + 3 + lane * 8;
  o[0] = A.x; o[1] = A.y; o[2] = A.z; o[3] = A.w;
  o[4] = B.x; o[5] = B.y; o[6] = B.z; o[7] = B.w;
  if (lane < 3) orow[lane] = mem_points[row * 3 + lane];
}

// ============================================================================
extern "C" void kernel_launch(void* const* d_in, const int* in_sizes, int n_in,
                              void* d_out, int out_size, void* d_ws, size_t ws_size,
                              hipStream_t stream) {
  const float* points     = (const float*)d_in[0];
  // d_in[1] (query descriptors) is unused by the reference's outputs
  const float* mem_points = (const float*)d_in[2];
  const float* mem_desc   = (const float*)d_in[3];
  const int*   usage      = (const int*)d_in[4];

  float* out       = (float*)d_out;
  float* out_usage = out + (size_t)NPTS * (FDIM + 3);
  float* out_idx   = out_usage + NPTS;

  char* ws = (char*)d_ws;
  float* pval  = (float*)(ws + OFF_PVAL);
  int*   pidx  = (int*)(ws + OFF_PIDX);
  int*   h1    = (int*)(ws + OFF_H1);
  int*   h2    = (int*)(ws + OFF_H2);
  int*   meta  = (int*)(ws + OFF_META);
  int*   buf   = (int*)(ws + OFF_BUF);
  int*   evict = (int*)(ws + OFF_EVICT);
  int*   delta = (int*)(ws + OFF_DELTA);

  // zero scratch that is read-modify-written (graph-capturable async memsets)
  hipMemsetAsync(h1, 0, 1024 * sizeof(int), stream);
  hipMemsetAsync(h2, 0, 512 * sizeof(int), stream);
  hipMemsetAsync(meta, 0, 8 * sizeof(int), stream);
  hipMemsetAsync(delta, 0, (size_t)NPTS * sizeof(int), stream);

  // NN search (WMMA)
  nn_wmma_kernel<<<dim3(QBLK, NSLICE), 256, 0, stream>>>(points, mem_points, pval, pidx);

  // stable least-used selection
  const int TB = 256, GB = (NPTS + TB - 1) / TB;
  hist1_kernel<<<GB, TB, 0, stream>>>(usage, h1);
  scan1_kernel<<<1, 1024, 0, stream>>>(h1, meta);
  hist2_kernel<<<GB, TB, 0, stream>>>(usage, meta, h2);
  scan2_kernel<<<1, 512, 0, stream>>>(h2, meta);
  collect_kernel<<<GB, TB, 0, stream>>>(usage, meta, buf);
  select_kernel<<<1, 1024, 0, stream>>>(meta, buf, evict);

  // per-query finalize + usage deltas
  finalize_kernel<<<1, 1024, 0, stream>>>(points, pval, pidx, evict, out_idx, delta);

  // outputs
  usage_out_kernel<<<GB, TB, 0, stream>>>(usage, delta, out_usage);
  copy_out_kernel<<<(NPTS + 7) / 8, 256, 0, stream>>>(mem_points, mem_desc, out);
}